// LinearAttention_395136991249
// MI455X (gfx1250) — compile-verified
//
#include <hip/hip_runtime.h>

// ---------------------------------------------------------------------------
// Linear attention for MI455X (gfx1250, wave32).
//   KtV[bh] = Kf^T @ Vm  (64x64, reduced over N via split-K + f32 atomics)
//   X       = Qf  @ KtV  (4096x64 @ 64x64 per bh)
// Bandwidth-bound (~100MB @ 23.3TB/s); fp32 WMMA 16x16x4 keeps reference
// precision and its throughput is hidden under the memory roof.
// CDNA5 paths: v_wmma_f32_16x16x4_f32 + global_load_async_to_lds_b128
// (ASYNCcnt) for the KtV staging in phase 2.
// ---------------------------------------------------------------------------

typedef __attribute__((ext_vector_type(2))) float v2f;
typedef __attribute__((ext_vector_type(8))) float v8f;

#define BB 2
#define HH 12
#define NN 4096
#define DD 64
#define LDSP (DD + 8)             // row stride 72: half-wave bank shift = 16 -> disjoint banks, 16B aligned
#define SPLITS 32                 // split-K blocks per (b,h) in phase 1
#define ROWS_P1 (NN / SPLITS)     // 128 rows of N per phase-1 block
#define SUB 64                    // LDS-staged sub-chunk of N rows
#define SCALE 0.125f              // N^-0.25 for N=4096

__device__ __forceinline__ float eluplus(float x) {
    // elu(x)+1 : x>0 -> x+1 ; else exp(x)
    return x > 0.0f ? (x + 1.0f) : __expf(x);
}

// Async global->LDS copy, 16 bytes per lane (GV mode). Tracked by ASYNCcnt.
__device__ __forceinline__ void async_load_b128(unsigned lds_off, const void* gaddr) {
    asm volatile("global_load_async_to_lds_b128 %0, %1, off"
                 :
                 : "v"(lds_off), "v"(gaddr)
                 : "memory");
}

__device__ __forceinline__ void wait_asynccnt0() {
#if __has_builtin(__builtin_amdgcn_s_wait_asynccnt)
    __builtin_amdgcn_s_wait_asynccnt(0);
#else
    asm volatile("s_wait_asynccnt 0" ::: "memory");
#endif
}

__global__ void zero_kernel(float* __restrict__ p, int n) {
    int i = blockIdx.x * blockDim.x + threadIdx.x;
    if (i < n) p[i] = 0.0f;
}

// ---------------------------------------------------------------------------
// Phase 1: per (b,h,split) block accumulates Kf^T @ Vm over 128 rows of N.
// 8 waves/block, each wave owns 2 of the 16 16x16 output tiles (same d-tile,
// so the A fragment is shared between the two WMMAs per k-step).
// ---------------------------------------------------------------------------
__global__ __launch_bounds__(256) void ktv_kernel(const float* __restrict__ K,
                                                  const float* __restrict__ V,
                                                  const float* __restrict__ mask,
                                                  float* __restrict__ ktv) {
    __shared__ float sK[SUB][LDSP];
    __shared__ float sV[SUB][LDSP];

    const int bh    = blockIdx.x / SPLITS;
    const int split = blockIdx.x % SPLITS;
    const int b     = bh / HH;
    const int tid   = threadIdx.x;
    const int wave  = tid >> 5;
    const int lane  = tid & 31;
    const int half  = lane >> 4;        // wave32: halves select K groups
    const int l16   = lane & 15;

    // wave -> two output tiles (same 16-wide d-slice, two 16-wide e-slices)
    const int tile0 = wave * 2;
    const int d0  = (tile0 >> 2) * 16;
    const int e0a = (tile0 & 3) * 16;
    const int e0b = ((tile0 + 1) & 3) * 16;

    v8f c0 = {};
    v8f c1 = {};

    const int n_block = split * ROWS_P1;

    for (int sc = 0; sc < ROWS_P1 / SUB; ++sc) {
        const int n0 = n_block + sc * SUB;

        // cooperative load + feature-map transform into LDS
        {
            const int row  = tid >> 2;          // 4 threads per 64-float row
            const int colq = (tid & 3) * 16;    // 16 floats = 4x float4
            const float mval = mask[b * NN + (n0 + row)];
            const float km   = mval * SCALE;    // Kf gets mask * scale
            const float* gK = K + ((size_t)bh * NN + n0 + row) * DD + colq;
            const float* gV = V + ((size_t)bh * NN + n0 + row) * DD + colq;
            #pragma unroll
            for (int j = 0; j < 4; ++j) {
                float4 kv = ((const float4*)gK)[j];
                float4 vv = ((const float4*)gV)[j];
                sK[row][colq + 4 * j + 0] = eluplus(kv.x) * km;
                sK[row][colq + 4 * j + 1] = eluplus(kv.y) * km;
                sK[row][colq + 4 * j + 2] = eluplus(kv.z) * km;
                sK[row][colq + 4 * j + 3] = eluplus(kv.w) * km;
                sV[row][colq + 4 * j + 0] = vv.x * mval;
                sV[row][colq + 4 * j + 1] = vv.y * mval;
                sV[row][colq + 4 * j + 2] = vv.z * mval;
                sV[row][colq + 4 * j + 3] = vv.w * mval;
            }
        }
        __syncthreads();

        #pragma unroll
        for (int ks = 0; ks < SUB / 4; ++ks) {
            const int kn = ks * 4 + 2 * half;   // this half-wave's K rows
            v2f a, bA, bB;
            // A = Kf^T fragment: A[m][k] = Kf[n0+k][d0+m]  (ISA 16x4 A layout)
            a.x  = sK[kn + 0][d0 + l16];
            a.y  = sK[kn + 1][d0 + l16];
            // B = Vm fragments: B[k][e] = Vm[n0+k][e0+e]
            bA.x = sV[kn + 0][e0a + l16];
            bA.y = sV[kn + 1][e0a + l16];
            bB.x = sV[kn + 0][e0b + l16];
            bB.y = sV[kn + 1][e0b + l16];
            c0 = __builtin_amdgcn_wmma_f32_16x16x4_f32(false, a, false, bA,
                                                       (short)0, c0, false, false);
            c1 = __builtin_amdgcn_wmma_f32_16x16x4_f32(false, a, false, bB,
                                                       (short)0, c1, false, false);
        }
        __syncthreads();
    }

    // split-K reduction via f32 atomics into the L2-resident 64x64 accumulator
    float* dst = ktv + (size_t)bh * DD * DD;
    #pragma unroll
    for (int r = 0; r < 8; ++r) {
        const int row = d0 + r + 8 * half;      // C/D layout: M = r + 8*half
        atomicAdd(&dst[row * DD + e0a + l16], c0[r]);
        atomicAdd(&dst[row * DD + e0b + l16], c1[r]);
    }
}

// ---------------------------------------------------------------------------
// Phase 2: X[n,:] = Qf[n,:] @ KtV. Each block handles 128 rows; each wave a
// 16-row tile with 4 accumulators (full 64-wide output), K=64 in 16 steps.
// KtV is staged into LDS with async global->LDS b128 copies (ASYNCcnt).
// ---------------------------------------------------------------------------
__global__ __launch_bounds__(256) void out_kernel(const float* __restrict__ Q,
                                                  const float* __restrict__ ktv,
                                                  float* __restrict__ X) {
    __shared__ float sM[DD][LDSP];

    const int nblocks = NN / 128;               // 32
    const int bh     = blockIdx.x / nblocks;
    const int n_base = (blockIdx.x % nblocks) * 128;
    const int tid  = threadIdx.x;
    const int wave = tid >> 5;
    const int lane = tid & 31;
    const int half = lane >> 4;
    const int l16  = lane & 15;

    // stage KtV (16KB) into LDS via async copies: 256 threads x 4 x b128
    {
        const int row  = tid >> 2;
        const int colq = (tid & 3) * 16;
        const float* g = ktv + (size_t)bh * DD * DD + row * DD + colq;
        // low 32 bits of the generic address = wave-relative LDS byte offset
        const unsigned lbase = (unsigned)(size_t)(&sM[row][colq]);
        #pragma unroll
        for (int j = 0; j < 4; ++j)
            async_load_b128(lbase + 16u * j, g + 4 * j);
        wait_asynccnt0();
    }
    __syncthreads();

    // preload this wave's 16 Q rows as A fragments (each element read once)
    const int qrow = n_base + wave * 16 + l16;
    const float* gQ = Q + ((size_t)bh * NN + qrow) * DD;
    v2f a[16];
    #pragma unroll
    for (int s = 0; s < 16; ++s) {
        float2 q = ((const float2*)gQ)[s * 2 + half]; // cols 4s+2*half .. +1
        a[s].x = eluplus(q.x) * SCALE;
        a[s].y = eluplus(q.y) * SCALE;
    }

    v8f c[4] = {v8f{}, v8f{}, v8f{}, v8f{}};
    #pragma unroll
    for (int s = 0; s < 16; ++s) {
        const int kn = s * 4 + 2 * half;
        #pragma unroll
        for (int e = 0; e < 4; ++e) {
            v2f bf;
            bf.x = sM[kn + 0][e * 16 + l16];
            bf.y = sM[kn + 1][e * 16 + l16];
            c[e] = __builtin_amdgcn_wmma_f32_16x16x4_f32(false, a[s], false, bf,
                                                         (short)0, c[e], false, false);
        }
    }

    // store D: row = r + 8*half, col = l16 (coalesced 64B per half-wave)
    #pragma unroll
    for (int e = 0; e < 4; ++e) {
        #pragma unroll
        for (int r = 0; r < 8; ++r) {
            const int orow = n_base + wave * 16 + r + 8 * half;
            X[((size_t)bh * NN + orow) * DD + e * 16 + l16] = c[e][r];
        }
    }
}

// ---------------------------------------------------------------------------

extern "C" void kernel_launch(void* const* d_in, const int* in_sizes, int n_in,
                              void* d_out, int out_size, void* d_ws, size_t ws_size,
                              hipStream_t stream) {
    const float* Q    = (const float*)d_in[0];
    const float* K    = (const float*)d_in[1];
    const float* V    = (const float*)d_in[2];
    const float* mask = (const float*)d_in[3];
    float* ktv = (float*)d_ws;   // B*H*64*64 floats = 384 KB
    float* X   = (float*)d_out;

    const int nktv = BB * HH * DD * DD;
    zero_kernel<<<(nktv + 255) / 256, 256, 0, stream>>>(ktv, nktv);
    ktv_kernel<<<BB * HH * SPLITS, 256, 0, stream>>>(K, V, mask, ktv);
    out_kernel<<<BB * HH * (NN / 128), 256, 0, stream>>>(Q, ktv, X);
}